// GAT_3083786519228
// MI455X (gfx1250) — compile-verified
//
#include <hip/hip_runtime.h>
#include <math.h>

typedef float v2f __attribute__((ext_vector_type(2)));
typedef float v8f __attribute__((ext_vector_type(8)));

#define TPB 256

// ---------------------------------------------------------------------------
// sign-aware float atomic max via integer atomics (always available)
__device__ __forceinline__ void atomicMaxF32(float* addr, float val) {
  if (val >= 0.0f) atomicMax((int*)addr, __float_as_int(val));
  else             atomicMin((unsigned int*)addr, __float_as_uint(val));
}

__global__ void fill_f32(float* __restrict__ p, float v, long n) {
  long i = (long)blockIdx.x * blockDim.x + threadIdx.x;
  if (i < n) p[i] = v;
}

// ---------------------------------------------------------------------------
// H[N, Cout] = X[N, Cin] @ W[Cout, Cin]^T  using V_WMMA_F32_16X16X4_F32.
// Each wave owns a (16*MT) x (16*NT) output tile, looping K in steps of 4.
// MT=2/NT=4 -> 64 accumulator VGPRs: fits the register file with no spills.
// All A/B fragments for a K-step are loaded BEFORE the WMMA block so the
// scheduler can clause the loads and wait once, instead of load->wait->wmma
// chains per B tile.
// Row tails are handled by CLAMPING the load row to N-1 (no lane divergence
// in the hot loop; garbage rows are simply never stored).
// A (16x4 f32): lanes 0-15 -> M=lane, VGPR j = K=j ; lanes 16-31 -> K=2+j.
// B (4x16 f32): lane n in [0,15] VGPR j = B[K=j][N=n]; lanes 16-31 K=2+j.
// C/D (16x16 f32): elem e of lane l -> M = e + (l>>4)*8, N = l&15.
template <int MT, int NT>
__global__ __launch_bounds__(TPB, 1)
void gemm_xwt_wmma(const float* __restrict__ X, const float* __restrict__ W,
                   float* __restrict__ Hout, int N, int Cin, int Cout) {
  const int wave = (int)(((long)blockIdx.x * blockDim.x + threadIdx.x) >> 5);
  const int lane = threadIdx.x & 31;
  const int colGroups = Cout / (16 * NT);
  const int tileM = wave / colGroups;
  const int cg    = wave - tileM * colGroups;
  const int row0  = tileM * (16 * MT);
  if (row0 >= N) return;               // uniform per wave: EXEC stays all-ones
  const int col0 = cg * (16 * NT);

  const int m  = lane & 15;
  const int kh = (lane >> 4) << 1;     // 0 or 2: which K-pair this half-wave holds

  v8f zero8 = {0.f,0.f,0.f,0.f,0.f,0.f,0.f,0.f};
  v8f acc[MT][NT];
#pragma unroll
  for (int i = 0; i < MT; ++i)
#pragma unroll
    for (int j = 0; j < NT; ++j) acc[i][j] = zero8;

  const float* xp[MT];
#pragma unroll
  for (int i = 0; i < MT; ++i) {
    int r = row0 + i * 16 + m;
    if (r > N - 1) r = N - 1;          // clamp: keeps loads valid, no divergence
    xp[i] = X + (size_t)r * Cin + kh;
  }
  const float* wp[NT];
#pragma unroll
  for (int j = 0; j < NT; ++j) wp[j] = W + (size_t)(col0 + j * 16 + m) * Cin + kh;

  for (int k = 0; k < Cin; k += 4) {
    v2f a[MT], b[NT];
#pragma unroll
    for (int i = 0; i < MT; ++i) a[i] = *(const v2f*)(xp[i] + k);
#pragma unroll
    for (int j = 0; j < NT; ++j) b[j] = *(const v2f*)(wp[j] + k);
#pragma unroll
    for (int j = 0; j < NT; ++j)
#pragma unroll
      for (int i = 0; i < MT; ++i)
        acc[i][j] = __builtin_amdgcn_wmma_f32_16x16x4_f32(false, a[i], false, b[j],
                                                          (short)0, acc[i][j], false, false);
  }

  const int n  = lane & 15;
  const int mr = (lane >> 4) << 3;
#pragma unroll
  for (int i = 0; i < MT; ++i)
#pragma unroll
    for (int j = 0; j < NT; ++j)
#pragma unroll
      for (int e = 0; e < 8; ++e) {
        int r = row0 + i * 16 + mr + e;
        if (r < N) Hout[(size_t)r * Cout + col0 + j * 16 + n] = acc[i][j][e];
      }
}

// ---------------------------------------------------------------------------
// s[n,h] = sum_c h[n,h,c]*a_src[h,c] ; d likewise. One wave per (node,head).
__global__ void gat_node_sd(const float* __restrict__ hf, const float* __restrict__ a_s,
                            const float* __restrict__ a_d, float* __restrict__ s,
                            float* __restrict__ d, int N, int H, int C) {
  const int wave = (int)(((long)blockIdx.x * blockDim.x + threadIdx.x) >> 5);
  const int lane = threadIdx.x & 31;
  if (wave >= N * H) return;
  const int node = wave / H, head = wave - node * H;
  const float* hp = hf + (size_t)node * H * C + (size_t)head * C;
  const float* as = a_s + head * C;
  const float* ad = a_d + head * C;
  float ss = 0.f, dd = 0.f;
  for (int c = lane; c < C; c += 32) { float v = hp[c]; ss += v * as[c]; dd += v * ad[c]; }
#pragma unroll
  for (int off = 16; off; off >>= 1) { ss += __shfl_down(ss, off, 32); dd += __shfl_down(dd, off, 32); }
  if (lane == 0) { s[wave] = ss; d[wave] = dd; }
}

// ---------------------------------------------------------------------------
// pass 1: segment max of leaky_relu(s[src]+d[dst], 0.2) over dst
__global__ void gat_edge_max(const int* __restrict__ src, const int* __restrict__ dst,
                             const float* __restrict__ s, const float* __restrict__ d,
                             float* __restrict__ emax, int E, int N, int H) {
  const int i = blockIdx.x * blockDim.x + threadIdx.x;
  if (i >= E + N) return;
  const int si = (i < E) ? src[i] : (i - E);
  const int di = (i < E) ? dst[i] : (i - E);
  for (int h = 0; h < H; ++h) {
    float e = s[si * H + h] + d[di * H + h];
    e = (e > 0.f) ? e : 0.2f * e;
    atomicMaxF32(&emax[di * H + h], e);
  }
}

// pass 2: p = exp(e - emax[dst]); store p, segment-sum into denom
__global__ void gat_edge_exp(const int* __restrict__ src, const int* __restrict__ dst,
                             const float* __restrict__ s, const float* __restrict__ d,
                             const float* __restrict__ emax, float* __restrict__ alpha,
                             float* __restrict__ denom, int E, int N, int H) {
  const int i = blockIdx.x * blockDim.x + threadIdx.x;
  if (i >= E + N) return;
  const int si = (i < E) ? src[i] : (i - E);
  const int di = (i < E) ? dst[i] : (i - E);
  for (int h = 0; h < H; ++h) {
    float e = s[si * H + h] + d[di * H + h];
    e = (e > 0.f) ? e : 0.2f * e;
    float p = expf(e - emax[di * H + h]);
    alpha[(size_t)i * H + h] = p;
    atomicAdd(&denom[di * H + h], p);
  }
}

// pass 3: out[dst,h,:] += (p/denom) * h[src,h,:]  — one wave per (edge, head)
__global__ void gat_edge_scatter(const int* __restrict__ src, const int* __restrict__ dst,
                                 const float* __restrict__ hf, const float* __restrict__ alpha,
                                 const float* __restrict__ denom, float* __restrict__ out,
                                 int E, int N, int H, int C) {
  const long wave = ((long)blockIdx.x * blockDim.x + threadIdx.x) >> 5;
  const int lane = threadIdx.x & 31;
  const long EN = (long)E + N;
  if (wave >= EN * H) return;
  const int i = (int)(wave / H);
  const int h = (int)(wave - (long)i * H);
  const int si = (i < E) ? src[i] : (i - E);
  const int di = (i < E) ? dst[i] : (i - E);
  const float w = alpha[(size_t)i * H + h] / (denom[di * H + h] + 1e-16f);
  const float* hp = hf + (size_t)si * H * C + (size_t)h * C;
  float* op = out + (size_t)di * H * C + (size_t)h * C;
  for (int c = lane; c < C; c += 32) atomicAdd(&op[c], w * hp[c]);
}

// ---------------------------------------------------------------------------
// in-place: y = leaky_relu(LN(y + bias)*g + b, 0.01). One wave per node.
__global__ void gat_ln_act(float* __restrict__ y, const float* __restrict__ bias,
                           const float* __restrict__ g, const float* __restrict__ b,
                           int N, int F, float slope) {
  const int wave = (int)(((long)blockIdx.x * blockDim.x + threadIdx.x) >> 5);
  const int lane = threadIdx.x & 31;
  if (wave >= N) return;
  float* row = y + (size_t)wave * F;
  float sum = 0.f, sq = 0.f;
  for (int c = lane; c < F; c += 32) { float v = row[c] + bias[c]; sum += v; sq += v * v; }
#pragma unroll
  for (int off = 16; off; off >>= 1) { sum += __shfl_xor(sum, off, 32); sq += __shfl_xor(sq, off, 32); }
  const float mu = sum / (float)F;
  const float var = sq / (float)F - mu * mu;
  const float rstd = rsqrtf(var + 1e-5f);
  for (int c = lane; c < F; c += 32) {
    float v = (row[c] + bias[c] - mu) * rstd * g[c] + b[c];
    row[c] = (v > 0.f) ? v : slope * v;
  }
}

// final: LN over 32 classes then log_softmax. One wave per node (lane == class).
__global__ void gat_final_ln_lsm(const float* __restrict__ agg, const float* __restrict__ bias,
                                 const float* __restrict__ g, const float* __restrict__ b,
                                 float* __restrict__ out, int N) {
  const int wave = (int)(((long)blockIdx.x * blockDim.x + threadIdx.x) >> 5);
  const int lane = threadIdx.x & 31;
  if (wave >= N) return;
  float v = agg[(size_t)wave * 32 + lane] + bias[lane];
  float sum = v;
#pragma unroll
  for (int off = 16; off; off >>= 1) sum += __shfl_xor(sum, off, 32);
  const float mu = sum * (1.0f / 32.0f);
  const float dv = v - mu;
  float sq = dv * dv;
#pragma unroll
  for (int off = 16; off; off >>= 1) sq += __shfl_xor(sq, off, 32);
  const float rstd = rsqrtf(sq * (1.0f / 32.0f) + 1e-5f);
  float y = dv * rstd * g[lane] + b[lane];
  float mx = y;
#pragma unroll
  for (int off = 16; off; off >>= 1) mx = fmaxf(mx, __shfl_xor(mx, off, 32));
  float ex = expf(y - mx);
  float se = ex;
#pragma unroll
  for (int off = 16; off; off >>= 1) se += __shfl_xor(se, off, 32);
  out[(size_t)wave * 32 + lane] = y - mx - logf(se);
}

// ---------------------------------------------------------------------------
extern "C" void kernel_launch(void* const* d_in, const int* in_sizes, int n_in,
                              void* d_out, int out_size, void* d_ws, size_t ws_size,
                              hipStream_t stream) {
  const float* x   = (const float*)d_in[0];
  const int*  esrc = (const int*) d_in[1];
  const int*  edst = (const int*) d_in[2];
  const float* W1  = (const float*)d_in[3];
  const float* as1 = (const float*)d_in[4];
  const float* ad1 = (const float*)d_in[5];
  const float* b1  = (const float*)d_in[6];
  const float* g1  = (const float*)d_in[7];
  const float* be1 = (const float*)d_in[8];
  const float* W2  = (const float*)d_in[9];
  const float* as2 = (const float*)d_in[10];
  const float* ad2 = (const float*)d_in[11];
  const float* b2  = (const float*)d_in[12];
  const float* g2  = (const float*)d_in[13];
  const float* be2 = (const float*)d_in[14];
  const float* W3  = (const float*)d_in[15];
  const float* as3 = (const float*)d_in[16];
  const float* ad3 = (const float*)d_in[17];
  const float* b3  = (const float*)d_in[18];
  const float* g3  = (const float*)d_in[19];
  const float* be3 = (const float*)d_in[20];
  float* out = (float*)d_out;

  const int N = 50000, E = 400000, EN = E + N;

  char* wsp = (char*)d_ws;
  auto carve = [&](size_t bytes) -> float* {
    float* p = (float*)wsp;
    wsp += (bytes + 255) & ~(size_t)255;
    return p;
  };
  float* bufH   = carve((size_t)N * 1024 * sizeof(float)); // projected features h
  float* bufAgg = carve((size_t)N * 1024 * sizeof(float)); // aggregated output / next input
  float* sb     = carve((size_t)N * 4 * sizeof(float));
  float* db     = carve((size_t)N * 4 * sizeof(float));
  float* emax   = carve((size_t)N * 4 * sizeof(float));
  float* denom  = carve((size_t)N * 4 * sizeof(float));
  float* alpha  = carve((size_t)EN * 4 * sizeof(float));

  auto run_layer = [&](const float* in, int Cin, const float* W, int Cout,
                       int Hh, int Cc, const float* a_s, const float* a_d) {
    // 1) projection H = in @ W^T (fp32 WMMA), 32-row tiles (MT=2, no spills)
    const int tilesM = (N + 31) / 32;
    if (Cout % 64 == 0) {
      long waves = (long)tilesM * (Cout / 64);
      gemm_xwt_wmma<2, 4><<<(int)((waves * 32 + TPB - 1) / TPB), TPB, 0, stream>>>(in, W, bufH, N, Cin, Cout);
    } else { // Cout == 32
      long waves = (long)tilesM * (Cout / 32);
      gemm_xwt_wmma<2, 2><<<(int)((waves * 32 + TPB - 1) / TPB), TPB, 0, stream>>>(in, W, bufH, N, Cin, Cout);
    }
    // 2) per-node attention scalars
    {
      long waves = (long)N * Hh;
      gat_node_sd<<<(int)((waves * 32 + TPB - 1) / TPB), TPB, 0, stream>>>(bufH, a_s, a_d, sb, db, N, Hh, Cc);
    }
    // 3) init accumulators (in is fully consumed by GEMM; stream-ordered reuse of bufAgg is safe)
    {
      long nsh = (long)N * Hh;
      fill_f32<<<(int)((nsh + TPB - 1) / TPB), TPB, 0, stream>>>(emax, -3.402823466e38f, nsh);
      fill_f32<<<(int)((nsh + TPB - 1) / TPB), TPB, 0, stream>>>(denom, 0.f, nsh);
      long nf = (long)N * Hh * Cc;
      fill_f32<<<(int)((nf + TPB - 1) / TPB), TPB, 0, stream>>>(bufAgg, 0.f, nf);
    }
    // 4) segment softmax + weighted scatter
    {
      int eb = (EN + TPB - 1) / TPB;
      gat_edge_max<<<eb, TPB, 0, stream>>>(esrc, edst, sb, db, emax, E, N, Hh);
      gat_edge_exp<<<eb, TPB, 0, stream>>>(esrc, edst, sb, db, emax, alpha, denom, E, N, Hh);
      long waves = (long)EN * Hh;
      gat_edge_scatter<<<(int)((waves * 32 + TPB - 1) / TPB), TPB, 0, stream>>>(
          esrc, edst, bufH, alpha, denom, bufAgg, E, N, Hh, Cc);
    }
  };

  const int lnBlocks = (int)(((long)N * 32 + TPB - 1) / TPB);

  // layer 1: 256 -> 4x256 concat
  run_layer(x, 256, W1, 1024, 4, 256, as1, ad1);
  gat_ln_act<<<lnBlocks, TPB, 0, stream>>>(bufAgg, b1, g1, be1, N, 1024, 0.01f);
  // layer 2: 1024 -> 4x128 concat
  run_layer(bufAgg, 1024, W2, 512, 4, 128, as2, ad2);
  gat_ln_act<<<lnBlocks, TPB, 0, stream>>>(bufAgg, b2, g2, be2, N, 512, 0.01f);
  // layer 3: 512 -> 32 (single head, mean == identity)
  run_layer(bufAgg, 512, W3, 32, 1, 32, as3, ad3);
  gat_final_ln_lsm<<<lnBlocks, TPB, 0, stream>>>(bufAgg, b3, g3, be3, out, N);
}